// LstmModel_11879879543488
// MI455X (gfx1250) — compile-verified
//
#include <hip/hip_runtime.h>

typedef float v2f __attribute__((ext_vector_type(2)));
typedef float v8f __attribute__((ext_vector_type(8)));

#define B_ 4096
#define T_ 2048
#define I_ 4
#define H_ 3
#define PD_ 32   // x-prefetch distance in timesteps (32*16B = 512B ahead)

__device__ __forceinline__ float fsig(float x) {
    // sigmoid(x) = 1 / (1 + 2^(-x*log2(e)))
    return __builtin_amdgcn_rcpf(1.0f + __builtin_amdgcn_exp2f(x * -1.44269504088896f));
}
__device__ __forceinline__ float ftanh(float x) {
    // tanh(x) = 2/(1 + 2^(-2x*log2(e))) - 1
    float e = __builtin_amdgcn_exp2f(x * -2.88539008177793f);
    return 2.0f * __builtin_amdgcn_rcpf(1.0f + e) - 1.0f;
}

// One wave32 per block; each wave owns 16 sequences (matrix columns).
// Gate rows reordered to (i0,f0,g0,o0, i1,f1,g1,o1, i2,f2,g2,o2, 0,0,0,0)
// so the D-matrix lands as: lane n (lo) -> units 0,1 of batch n in VGPR0..7,
//                           lane n (hi) -> unit 2 of batch n in VGPR0..3.
__global__ __launch_bounds__(32) void lstm_wmma_kernel(
    const float* __restrict__ x,    // [B,T,4]
    const float* __restrict__ Wih,  // [12,4] rows: i0,i1,i2,f0,f1,f2,g0,g1,g2,o0,o1,o2
    const float* __restrict__ Whh,  // [12,3]
    const float* __restrict__ bih,  // [12]
    const float* __restrict__ bhh,  // [12]
    const int*   __restrict__ len,  // [B]
    float* __restrict__ out)        // [B,T,3]
{
    const int  lane = threadIdx.x;      // 0..31
    const int  n    = lane & 15;        // batch-in-tile == matrix column == A row
    const bool hi   = lane >= 16;
    const int  b    = blockIdx.x * 16 + n;

    // ---- A-matrix layout (16x4 f32): lane m holds row m; VGPR0 = K{0|2}, VGPR1 = K{1|3}
    // desired row r: unit j = r>>2, gate k = r&3  ->  source row = k*3 + j
    v2f aWih = {0.f, 0.f}, aWhh = {0.f, 0.f};
    {
        const int src = (n & 3) * 3 + (n >> 2);
        if (n < 12) {
            aWih.x = Wih[src * 4 + (hi ? 2 : 0)];
            aWih.y = Wih[src * 4 + (hi ? 3 : 1)];
            aWhh.x = Whh[src * 3 + (hi ? 2 : 0)];   // K=3 padded with 0
            aWhh.y = hi ? 0.f : Whh[src * 3 + 1];
        }
    }

    // ---- C-matrix bias: VGPR r = row (hi ? 8+r : r), broadcast across columns
    v8f cbias;
#pragma unroll
    for (int r = 0; r < 8; ++r) {
        const int row = hi ? 8 + r : r;
        float v = 0.f;
        if (row < 12) {
            const int s = (row & 3) * 3 + (row >> 2);
            v = bih[s] + bhh[s];
        }
        cbias[r] = v;
    }

    const int mylen = len[b];

    // x B-matrix (4x16): VGPR0 = K{0|2} col n, VGPR1 = K{1|3} col n -> one b64/lane
    const float* xp  = x   + (size_t)b * T_ * I_ + (hi ? 2 : 0);
    float*       op0 = out + (size_t)b * T_ * H_ + (hi ? 2 : 0);
    float*       op1 = out + (size_t)b * T_ * H_ + (hi ? 2 : 1);

    float cA = 0.f, hA = 0.f, cB = 0.f, hB = 0.f;

    // one LSTM step on the already-loaded x operand
    auto step = [&](int t, v2f xv) __attribute__((always_inline)) {
        // h B-matrix: VGPR0 = h0(lo)/h2(hi) == hA everywhere; VGPR1 = h1(lo)/0(hi)
        v2f bH;
        bH.x = hA;
        bH.y = hi ? 0.f : hB;

        v8f acc = __builtin_amdgcn_wmma_f32_16x16x4_f32(
            false, aWih, false, xv, (short)0, cbias, false, false);
        v8f g = __builtin_amdgcn_wmma_f32_16x16x4_f32(
            false, aWhh, false, bH, (short)0, acc, false, false);

        // unit A: rows r..r+3 = (i,f,g,o); lo lanes = unit0, hi lanes = unit2
        const float iA = fsig(g[0]), fA = fsig(g[1]), gA = ftanh(g[2]), oA = fsig(g[3]);
        cA = fA * cA + iA * gA;
        hA = oA * ftanh(cA);
        // unit B: lo lanes = unit1; hi lanes see exact zeros -> hB stays 0
        const float iB = fsig(g[4]), fB = fsig(g[5]), gB = ftanh(g[6]), oB = fsig(g[7]);
        cB = fB * cB + iB * gB;
        hB = oB * ftanh(cB);

        // masked store; hi lanes write h2 twice (same value) to keep EXEC all-ones
        const bool  valid = (t < mylen);
        const float s0 = valid ? hA : 0.f;              // lo: h0, hi: h2
        const float s1 = valid ? (hi ? hA : hB) : 0.f;  // lo: h1, hi: h2 (dup)
        op0[(size_t)t * H_] = s0;
        op1[(size_t)t * H_] = s1;
    };

    v2f xcur = *(const v2f*)xp;                 // t = 0
    int t = 0;
    // steady state: unconditional next-step load + deep line prefetch (in-bounds)
    for (; t < T_ - 1 - PD_; ++t) {
        __builtin_prefetch(xp + (size_t)PD_ * I_, 0, 0);   // global_prefetch_b8, 512B ahead
        xp += I_;
        v2f xn = *(const v2f*)xp;
        step(t, xcur);
        xcur = xn;
    }
    // tail: no prefetch (would run past this row), still double-buffered
    for (; t < T_ - 1; ++t) {
        xp += I_;
        v2f xn = *(const v2f*)xp;
        step(t, xcur);
        xcur = xn;
    }
    step(T_ - 1, xcur);                         // peeled last step: no load at all
}

extern "C" void kernel_launch(void* const* d_in, const int* in_sizes, int n_in,
                              void* d_out, int out_size, void* d_ws, size_t ws_size,
                              hipStream_t stream) {
    (void)in_sizes; (void)n_in; (void)d_ws; (void)ws_size; (void)out_size;
    const float* x    = (const float*)d_in[0];
    const float* Wih  = (const float*)d_in[1];
    const float* Whh  = (const float*)d_in[2];
    const float* bih  = (const float*)d_in[3];
    const float* bhh  = (const float*)d_in[4];
    const int*   len  = (const int*)d_in[5];
    float* out = (float*)d_out;

    dim3 grid(B_ / 16);   // 256 waves, one per block for maximal WGP spread
    dim3 block(32);
    lstm_wmma_kernel<<<grid, block, 0, stream>>>(x, Wih, Whh, bih, bhh, len, out);
}